// DAGrid_74680891343626
// MI455X (gfx1250) — compile-verified
//
#include <hip/hip_runtime.h>
#include <math.h>

#define NLVL 8
#define OUTW 51
#define BLK  256

struct F3 { float x, y, z; };                       // 12B anchor entry
typedef float        v4f  __attribute__((ext_vector_type(4)));
typedef unsigned int u32x4 __attribute__((ext_vector_type(4)));
typedef int          i32x4 __attribute__((ext_vector_type(4)));
typedef int          i32x8 __attribute__((ext_vector_type(8)));

#if defined(__has_builtin)
#if __has_builtin(__builtin_amdgcn_tensor_store_from_lds)
#define USE_TDM 1
#endif
#endif
#ifndef USE_TDM
#define USE_TDM 0
#endif

__global__ __launch_bounds__(BLK) void dagrid_embed_kernel(
    const float* __restrict__ xyz,
    const float* __restrict__ data,
    const int*   __restrict__ alpha_ratio,
    float*       __restrict__ out,
    int npts)
{
    __shared__ __align__(16) float smem[BLK * OUTW];

    const int tid = threadIdx.x;
    const int n   = blockIdx.x * BLK + tid;

    // alpha_ratio is a uniform scalar input -> s_load
    const float ar = fminf((float)(*alpha_ratio), 1.0f);

    if (n < npts) {
        // streaming read-once inputs: non-temporal so the anchor table keeps L2
        const float px = __builtin_nontemporal_load(xyz + 3 * n + 0);
        const float py = __builtin_nontemporal_load(xyz + 3 * n + 1);
        const float pz = __builtin_nontemporal_load(xyz + 3 * n + 2);
        smem[tid * OUTW + 0] = px;   // passthrough of ORIGINAL xyz
        smem[tid * OUTW + 1] = py;
        smem[tid * OUTW + 2] = pz;

        const float HI = 1.0f - 1e-6f;                 // bounds_hi after eps
        const float xn = (fminf(fmaxf(px, -1.0f), HI) + 1.0f) * 0.5f;  // /SIZE==2
        const float yn = (fminf(fmaxf(py, -1.0f), HI) + 1.0f) * 0.5f;
        const float zn = (fminf(fmaxf(pz, -1.0f), HI) + 1.0f) * 0.5f;

        constexpr int kScale[NLVL] = {16, 21, 28, 39, 52, 70, 95, 128};
        constexpr int kOff  [NLVL] = {0, 4913, 15561, 39950, 103950,
                                      252827, 610738, 1495474};

        #pragma unroll
        for (int l = 0; l < NLVL; ++l) {
            const int   s  = kScale[l];
            const int   r1 = s + 1;
            const float sf = (float)s;
            const float fr = (float)(1 << l);          // frequency 2^l

            const float fx = xn * sf, fy = yn * sf, fz = zn * sf;
            const int ix0 = (int)fx, iy0 = (int)fy, iz0 = (int)fz;
            // faithful to reference: per-corner float add, THEN truncate
            const int ix1 = (int)(fx + 1.0f);
            const int iy1 = (int)(fy + 1.0f);
            const int iz1 = (int)(fz + 1.0f);
            const float ox = fx - (float)ix0;
            const float oy = fy - (float)iy0;
            const float oz = fz - (float)iz0;

            const int r2  = r1 * r1;
            const int xo0 = ix0 * r2 + kOff[l];
            const int xo1 = ix1 * r2 + kOff[l];
            const int yo0 = iy0 * r1;
            const int yo1 = iy1 * r1;

            float a0 = 0.f, a1 = 0.f, a2 = 0.f, a3 = 0.f, a4 = 0.f, a5 = 0.f;
            #pragma unroll
            for (int c = 0; c < 8; ++c) {
                const int cx = (c >> 2) & 1, cy = (c >> 1) & 1, cz = c & 1;
                const int idx = (cx ? xo1 : xo0) + (cy ? yo1 : yo0)
                              + (cz ? iz1 : iz0);
                // 12B gather; table stays L2-resident (default RT temporal hint)
                const F3 d = *(const F3*)(data + 3 * idx);
                const float w = (cx ? ox : 1.0f - ox)
                              * (cy ? oy : 1.0f - oy)
                              * (cz ? oz : 1.0f - oz);
                const float vx = d.x * fr, vy = d.y * fr, vz = d.z * fr;
                // |v| <= 128 << 256*pi: hardware v_sin/v_cos are valid here
                a0 += w * __sinf(vx);  a1 += w * __sinf(vy);  a2 += w * __sinf(vz);
                a3 += w * __cosf(vx);  a4 += w * __cosf(vy);  a5 += w * __cosf(vz);
            }

            // coarse-to-fine anneal weight (==1.0 when alpha_ratio>=1)
            const float t  = fminf(fmaxf(ar * (float)NLVL - (float)l, 0.0f), 1.0f);
            const float lw = (1.0f - __cosf(3.14159265358979323846f * t)) * 0.5f;

            float* o = &smem[tid * OUTW + 3 + l * 6];
            o[0] = a0 * lw; o[1] = a1 * lw; o[2] = a2 * lw;
            o[3] = a3 * lw; o[4] = a4 * lw; o[5] = a5 * lw;
        }
    }
    __syncthreads();

#if USE_TDM
    // ---- Tensor Data Mover writeback: one DMA of the whole 52KB block tile.
    // D# layout per CDNA5 ISA ch.8.3/8.4. 1-D tile: tile_dim0=BLK*OUTW dwords,
    // tensor_dim0 clamped to the remaining valid elements so TDM drops OOB
    // writes for a partial last block. Tracked on TENSORcnt; S_ENDPGM's
    // implicit wait-idle drains it before LDS is released.
    if (tid == 0) {
        const unsigned lds_base =
            (unsigned)(unsigned long long)(void*)smem;   // LDS aperture: low 32 bits = LDS byte offset
        const unsigned long long gaddr =
            (unsigned long long)(void*)out
            + (unsigned long long)blockIdx.x * (unsigned long long)(BLK * OUTW * 4);

        long long remain = (long long)npts * OUTW
                         - (long long)blockIdx.x * (BLK * OUTW);
        if (remain > BLK * OUTW) remain = BLK * OUTW;
        if (remain < 0)          remain = 0;
        const unsigned dim0 = (unsigned)remain;          // OOB clamp in elements

        u32x4 g0;
        g0[0] = 0x1u;                                    // count=1 (valid user D#)
        g0[1] = lds_base;                                // lds_addr [63:32]
        g0[2] = (unsigned)(gaddr & 0xFFFFFFFFu);         // global_addr[31:0]
        g0[3] = (unsigned)((gaddr >> 32) & 0x01FFFFFFu)  // global_addr[56:32]
              | (2u << 30);                              // type=2 ("image")

        i32x8 g1;
        g1[0] = (int)(2u << 16);                         // data_size=2 -> 4B elems
        g1[1] = (int)((dim0 & 0xFFFFu) << 16);           // tensor_dim0[15:0]  @ [79:48]
        g1[2] = (int)(((dim0 >> 16) & 0xFFFFu)           // tensor_dim0[31:16]
              | (1u << 16));                             // tensor_dim1=1      @ [111:80]
        g1[3] = (int)(((unsigned)(BLK * OUTW)) << 16);   // tile_dim0=13056    @ [127:112]
        g1[4] = 0;                                       // tile_dim1=0 (unused), tile_dim2=0
        g1[5] = (int)(BLK * OUTW);                       // tensor_dim0_stride[31:0]
        g1[6] = 0;                                       // stride0[47:32], stride1[15:0]
        g1[7] = 0;

        const i32x4 gz4 = {0, 0, 0, 0};                  // groups 2/3: unused (<=2D)
        const i32x8 gz8 = {0, 0, 0, 0, 0, 0, 0, 0};      // extra operand (clang-23 form)
        // therock/clang-23 6-arg form: (g0, g1, g2, g3, i32x8, cpol)
        __builtin_amdgcn_tensor_store_from_lds(g0, g1, gz4, gz4, gz8, 0 /*cpol*/);
    }
#else
    // ---- Fallback: coalesced non-temporal streaming writeback, float4 granules.
    const long long tot4   = ((long long)npts * OUTW) >> 2;
    const long long gbase4 = (long long)blockIdx.x * (BLK * OUTW / 4);
    const v4f* sv = (const v4f*)smem;
    #pragma unroll 1
    for (int i = tid; i < (BLK * OUTW) / 4; i += BLK) {
        const long long gi = gbase4 + i;
        if (gi < tot4)
            __builtin_nontemporal_store(sv[i], ((v4f*)out) + gi);
    }
    const long long totalF = (long long)npts * OUTW;
    const long long rem    = totalF - (tot4 << 2);
    if (rem > 0) {
        const long long g = (tot4 << 2) + tid;
        if (g < totalF) {
            const long long li = g - (long long)blockIdx.x * (BLK * OUTW);
            if (li >= 0 && li < BLK * OUTW)
                __builtin_nontemporal_store(smem[(int)li], out + g);
        }
    }
#endif
}

extern "C" void kernel_launch(void* const* d_in, const int* in_sizes, int n_in,
                              void* d_out, int out_size, void* d_ws, size_t ws_size,
                              hipStream_t stream)
{
    const float* xyz   = (const float*)d_in[0];   // [N,3] float32
    const float* data  = (const float*)d_in[1];   // [~3.64M,3] float32
    const int*   ar    = (const int*)d_in[2];     // scalar int
    float*       outp  = (float*)d_out;           // [N,51] float32
    const int    npts  = in_sizes[0] / 3;
    const int    blocks = (npts + BLK - 1) / BLK;
    dagrid_embed_kernel<<<blocks, BLK, 0, stream>>>(xyz, data, ar, outp, npts);
}